// GCNN_21337397526628
// MI455X (gfx1250) — compile-verified
//
#include <hip/hip_runtime.h>
#include <math.h>

#define NN 100000
#define EE 1600000
#define HH 100
#define FIN 128

typedef float v2f __attribute__((ext_vector_type(2)));
typedef float v8f __attribute__((ext_vector_type(8)));

// ---------------- degree / norm precompute ----------------

__global__ void deg_kernel(const int* __restrict__ dst, float* __restrict__ deg) {
    int e = blockIdx.x * blockDim.x + threadIdx.x;
    if (e < EE) atomicAdd(&deg[dst[e]], 1.0f);
}

__global__ void dinv_kernel(float* __restrict__ deg) {
    int i = blockIdx.x * blockDim.x + threadIdx.x;
    if (i < NN) deg[i] = rsqrtf(deg[i] + 1.0f);   // +1 for self loop
}

__global__ void norm_kernel(const int* __restrict__ src, const int* __restrict__ dst,
                            const float* __restrict__ dinv, float* __restrict__ norm) {
    int e = blockIdx.x * blockDim.x + threadIdx.x;
    if (e < EE) norm[e] = dinv[src[e]] * dinv[dst[e]];
}

__global__ void lsum_kernel(const float* __restrict__ labels, float* __restrict__ sums) {
    int i = blockIdx.x * blockDim.x + threadIdx.x;
    float v = (i < NN) ? labels[i] : 0.0f;
    #pragma unroll
    for (int off = 16; off > 0; off >>= 1) v += __shfl_down(v, off, 32);
    if ((threadIdx.x & 31) == 0) atomicAdd(&sums[0], v);
}

// ---------------- GEMM via V_WMMA_F32_16X16X4_F32 ----------------
// One wave computes a 16-row strip across all 112 (=7*16) padded output cols.
// A-frag layout (16x4 f32): lane m = lane&15 is row; kh = lane>>4 picks K pair
//   {k+2kh, k+2kh+1}. B-frag (4x16): VGPR0 = rows {k, k+2}, VGPR1 = {k+1, k+3}
//   by lane half; col = lane&15. C/D: acc[r] = OUT[row0 + r + 8*kh][col0 + m].

__global__ void gemm_kernel(const float* __restrict__ A, const float* __restrict__ W,
                            const float* __restrict__ bias, float* __restrict__ out,
                            int K) {
    int wave = blockIdx.x * (blockDim.x >> 5) + (threadIdx.x >> 5);
    int row0 = wave << 4;
    if (row0 >= NN) return;                 // whole-wave guard: EXEC stays all-1s
    int lane = threadIdx.x & 31;
    int m  = lane & 15;
    int kh = lane >> 4;

    v8f acc[7];
    #pragma unroll
    for (int t = 0; t < 7; t++) acc[t] = v8f{0,0,0,0,0,0,0,0};

#if __has_builtin(__builtin_amdgcn_wmma_f32_16x16x4_f32)
    const float* arow = A + (size_t)(row0 + m) * K;
    for (int k = 0; k < K; k += 4) {
        v2f a;
        a.x = arow[k + 2 * kh + 0];
        a.y = arow[k + 2 * kh + 1];
        #pragma unroll
        for (int t = 0; t < 7; t++) {
            int col = t * 16 + m;
            v2f b;
            if (col < HH) {
                b.x = W[(k + 2 * kh + 0) * HH + col];
                b.y = W[(k + 2 * kh + 1) * HH + col];
            } else {
                b.x = 0.0f; b.y = 0.0f;
            }
            acc[t] = __builtin_amdgcn_wmma_f32_16x16x4_f32(
                false, a, false, b, (short)0, acc[t], false, false);
        }
    }
#else
    // scalar fallback (keeps identical output layout)
    for (int k = 0; k < K; k++) {
        #pragma unroll
        for (int t = 0; t < 7; t++) {
            int col = t * 16 + m;
            float wv = (col < HH) ? W[k * HH + col] : 0.0f;
            #pragma unroll
            for (int r = 0; r < 8; r++)
                acc[t][r] += A[(size_t)(row0 + r + 8 * kh) * K + k] * wv;
        }
    }
#endif

    #pragma unroll
    for (int t = 0; t < 7; t++) {
        int col = t * 16 + m;
        if (col >= HH) continue;
        float bv = bias ? bias[col] : 0.0f;
        #pragma unroll
        for (int r = 0; r < 8; r++) {
            int row = row0 + r + 8 * kh;
            out[(size_t)row * HH + col] = acc[t][r] + bv;
        }
    }
}

// ---------------- aggregation ----------------
// agg = tmp * dinv^2 (self loop) + bias, then edge scatter-add on top.

__global__ void selfinit_kernel(const float* __restrict__ tmp, const float* __restrict__ dinv,
                                const float* __restrict__ bias, float* __restrict__ agg) {
    int idx = blockIdx.x * blockDim.x + threadIdx.x;
    if (idx < NN * HH) {
        int node = idx / HH;
        int f    = idx - node * HH;
        float dn = dinv[node];
        agg[idx] = tmp[idx] * dn * dn + bias[f];
    }
}

__global__ void scatter_kernel(const float* __restrict__ tmp, const float* __restrict__ norm,
                               const int* __restrict__ src, const int* __restrict__ dst,
                               float* __restrict__ agg) {
    int tid = blockIdx.x * blockDim.x + threadIdx.x;   // EE*25 = 40M fits in int32
    if (tid >= EE * 25) return;
    int e = tid / 25;
    int f = (tid - e * 25) * 4;
    int s = src[e], d = dst[e];
    float nv = norm[e];
    // row stride = 400 bytes (16B aligned), f*4 bytes is 16B aligned -> float4 OK
    float4 v = *reinterpret_cast<const float4*>(tmp + (size_t)s * HH + f);
    float* o = agg + (size_t)d * HH + f;
    atomicAdd(o + 0, v.x * nv);
    atomicAdd(o + 1, v.y * nv);
    atomicAdd(o + 2, v.z * nv);
    atomicAdd(o + 3, v.w * nv);
}

// ---------------- head: fc2 + sigmoid + weighted BCE ----------------

__global__ void head_kernel(const float* __restrict__ h, const float* __restrict__ w2,
                            const float* __restrict__ b2, const float* __restrict__ labels,
                            float* __restrict__ sums, float* __restrict__ out) {
    int i = blockIdx.x * blockDim.x + threadIdx.x;
    float contrib = 0.0f;
    if (i < NN) {
        float acc = b2[0];
        const float* hr = h + (size_t)i * HH;
        #pragma unroll 4
        for (int f = 0; f < HH; f++) acc += hr[f] * w2[f];
        float p = 1.0f / (1.0f + expf(-acc));
        out[1 + i] = p;
        float pc  = fminf(fmaxf(p, 1e-7f), 1.0f - 1e-7f);
        float lab = labels[i];
        float pos = sums[0];
        float w   = ((float)NN - pos) / pos * lab + 1.0f;
        contrib   = w * (-(lab * logf(pc) + (1.0f - lab) * logf(1.0f - pc)));
    }
    #pragma unroll
    for (int off = 16; off > 0; off >>= 1) contrib += __shfl_down(contrib, off, 32);
    if ((threadIdx.x & 31) == 0) atomicAdd(&sums[1], contrib);
}

__global__ void finalize_kernel(const float* __restrict__ sums, float* __restrict__ out) {
    if (threadIdx.x == 0 && blockIdx.x == 0) out[0] = sums[1] / (float)NN;
}

// ---------------- driver ----------------

extern "C" void kernel_launch(void* const* d_in, const int* in_sizes, int n_in,
                              void* d_out, int out_size, void* d_ws, size_t ws_size,
                              hipStream_t stream) {
    const float* x      = (const float*)d_in[0];
    const int*   ei     = (const int*)d_in[1];
    const float* labels = (const float*)d_in[2];
    const float* W1     = (const float*)d_in[3];
    const float* b1     = (const float*)d_in[4];
    const float* Wh     = (const float*)d_in[5];
    const float* bh     = (const float*)d_in[6];
    const float* fc1W   = (const float*)d_in[7];
    const float* fc1b   = (const float*)d_in[8];
    const float* fc2W   = (const float*)d_in[9];
    const float* fc2b   = (const float*)d_in[10];
    float* out = (float*)d_out;

    float* ws   = (float*)d_ws;
    float* dinv = ws;                        // NN
    float* sums = ws + NN;                   // 4 (label_sum, loss_acc, pad, pad)
    float* norm = sums + 4;                  // EE
    float* bufA = norm + EE;                 // NN*HH
    float* bufB = bufA + (size_t)NN * HH;    // NN*HH

    const int* src = ei;
    const int* dst = ei + EE;

    // zero deg + reduction accumulators
    hipMemsetAsync(dinv, 0, (size_t)(NN + 4) * sizeof(float), stream);

    deg_kernel <<<(EE + 255) / 256, 256, 0, stream>>>(dst, dinv);
    dinv_kernel<<<(NN + 255) / 256, 256, 0, stream>>>(dinv);
    norm_kernel<<<(EE + 255) / 256, 256, 0, stream>>>(src, dst, dinv, norm);
    lsum_kernel<<<(NN + 255) / 256, 256, 0, stream>>>(labels, sums);

    const int gemmBlocks    = (NN / 16 + 3) / 4;            // 4 waves per 128-thread block
    const int initBlocks    = (NN * HH + 255) / 256;
    const int scatterBlocks = (EE * 25 + 255) / 256;

    // layer 1: 128 -> 100
    gemm_kernel    <<<gemmBlocks, 128, 0, stream>>>(x, W1, nullptr, bufA, FIN);
    selfinit_kernel<<<initBlocks, 256, 0, stream>>>(bufA, dinv, b1, bufB);
    scatter_kernel <<<scatterBlocks, 256, 0, stream>>>(bufA, norm, src, dst, bufB);

    // hidden layers 2..9: 100 -> 100
    const float* h = bufB;
    for (int l = 0; l < 8; l++) {
        gemm_kernel    <<<gemmBlocks, 128, 0, stream>>>(h, Wh + (size_t)l * HH * HH,
                                                        nullptr, bufA, HH);
        selfinit_kernel<<<initBlocks, 256, 0, stream>>>(bufA, dinv, bh + (size_t)l * HH, bufB);
        scatter_kernel <<<scatterBlocks, 256, 0, stream>>>(bufA, norm, src, dst, bufB);
        h = bufB;
    }

    // fc1 (bias fused into GEMM store)
    gemm_kernel<<<gemmBlocks, 128, 0, stream>>>(h, fc1W, fc1b, bufA, HH);

    // fc2 + sigmoid + weighted BCE
    head_kernel    <<<(NN + 255) / 256, 256, 0, stream>>>(bufA, fc2W, fc2b, labels, sums, out);
    finalize_kernel<<<1, 32, 0, stream>>>(sums, out);

    (void)in_sizes; (void)n_in; (void)out_size; (void)ws_size;
}